// NonLocalModel_30691836297238
// MI455X (gfx1250) — compile-verified
//
#include <hip/hip_runtime.h>

typedef __attribute__((ext_vector_type(2))) float v2f;
typedef __attribute__((ext_vector_type(8))) float v8f;
typedef __attribute__((ext_vector_type(4))) float f4;

#define U_DIM 128
#define X2D   16
#define OUT_W 1024
#define WAVES 4
#define EPW   16                       // edges per wave (M tile)
#define EPB   (WAVES * EPW)            // 64 edges per block
#define PITCH 132                      // floats; conflict-free strided A-frag reads
#define KSTEPS 32                      // K/4
#define KS2    16                      // K/8 (two WMMA k-steps per packed float4)
#define PF     8                       // B prefetch depth (rotating buffer)
#define STG_FLOATS 1856                // 16 rows * (16*7+4) floats (worst case dim=7)
#define PACK_BYTES (16 * KS2 * 32 * 4 * sizeof(float))   // 131072

__device__ __forceinline__ v8f wmma4(v2f a, v2f b, v8f c) {
    return __builtin_amdgcn_wmma_f32_16x16x4_f32(false, a, false, b,
                                                 (short)0, c, false, false);
}

// ---- weight repack: pack[((nt*16 + ks2)*32 + lane)*4] = the 4 floats lane
//      needs for WMMA k-steps 2*ks2 and 2*ks2+1 of N-tile nt ----
__global__ __launch_bounds__(256)
void w_pack_kernel(const float* __restrict__ W, float* __restrict__ pack)
{
    const int gid = blockIdx.x * 256 + threadIdx.x;   // 0 .. 8191
    if (gid >= 16 * KS2 * 32) return;
    const int lane = gid & 31;
    const int ks2  = (gid >> 5) & 15;
    const int nt   = gid >> 9;
    const int b    = nt >> 2;
    const int col  = (nt & 3) * 16 + (lane & 15);
    const int kk   = (lane >> 4) * 2;
    const float* Wb = W + (size_t)b * (U_DIM * 64) + col;
    f4 v;
    v.x = Wb[(size_t)(ks2 * 8 + kk)     * 64];
    v.y = Wb[(size_t)(ks2 * 8 + kk + 1) * 64];
    v.z = Wb[(size_t)(ks2 * 8 + 4 + kk)     * 64];
    v.w = Wb[(size_t)(ks2 * 8 + 4 + kk + 1) * 64];
    *(f4*)(pack + (size_t)gid * 4) = v;
}

// ---- GEMM inner body: 16 packed f4 -> 32 WMMAs; affine prefetch (no clamp) ----
template <bool PREF>
__device__ __forceinline__
void gemm_tile(v8f& acc, const v2f (&af)[KSTEPS], f4 (&buf)[PF],
               const f4* __restrict__ Bp, int nt)
{
    #pragma unroll
    for (int ks2 = 0; ks2 < KS2; ++ks2) {
        const f4 bq = buf[ks2 & (PF - 1)];        // compile-time slot index
        v2f b0; b0.x = bq.x; b0.y = bq.y;
        v2f b1; b1.x = bq.z; b1.y = bq.w;
        acc = wmma4(af[2 * ks2],     b0, acc);
        acc = wmma4(af[2 * ks2 + 1], b1, acc);
        if (PREF)                                  // affine: base + imm offsets
            buf[ks2 & (PF - 1)] = Bp[(size_t)(nt * KS2 + ks2 + PF) * 32];
    }
}

__device__ __forceinline__
void gemm_tile_direct(v8f& acc, const v2f (&af)[KSTEPS],
                      const float* __restrict__ W, int nt, int llo, int kk)
{
    const float* Wp = W + (size_t)(nt >> 2) * (U_DIM * 64) + (nt & 3) * 16 + llo;
    #pragma unroll
    for (int ks = 0; ks < KSTEPS; ++ks) {
        v2f bf;
        bf.x = Wp[(size_t)(ks * 4 + kk)     * 64];
        bf.y = Wp[(size_t)(ks * 4 + kk + 1) * 64];
        acc = wmma4(af[ks], bf, acc);
    }
}

// ---- fused store: products -> LDS (conflict-free) -> coalesced b128 stores ----
template <int DIM>
__device__ __forceinline__
void store_tile(float* __restrict__ outrow0,      // out + e0*OUT_W + off + 16*ntl*DIM
                const float* __restrict__ x2p,    // &x2s[lm*X2D + s0]
                float* __restrict__ stg,          // wave-private staging (in x1s)
                v8f acc, int lane, int lhi, int llo)
{
    if (DIM == 1) {                               // already contiguous: direct b32
        #pragma unroll
        for (int r = 0; r < 8; ++r) {
            const int M = r + 8 * lhi;
            outrow0[(size_t)M * OUT_W + llo] = acc[r] * x2p[M * X2D];
        }
        return;
    }
    const int RS = 16 * DIM + 4;                  // staging row stride (floats)
    #pragma unroll
    for (int r = 0; r < 8; ++r) {
        const int   M  = r + 8 * lhi;
        const float cv = acc[r];
        float* sp = stg + M * RS + llo * DIM;
        #pragma unroll
        for (int m = 0; m < DIM; ++m)
            sp[m] = cv * x2p[M * X2D + m];
    }
    // readback: 16 rows x 4*DIM float4 each, coalesced aligned b128 stores
    #pragma unroll
    for (int it = 0; it < 2 * DIM; ++it) {
        const int f   = it * 32 + lane;           // 0 .. 64*DIM-1
        const int row = f / (4 * DIM);
        const int c4  = f - row * (4 * DIM);
        f4 v = *(const f4*)(stg + row * RS + c4 * 4);
        *(f4*)(outrow0 + (size_t)row * OUT_W + c4 * 4) = v;
    }
}

__device__ __forceinline__
void store_dispatch(int nt, v8f acc, float* __restrict__ out,
                    const float* __restrict__ x2s_, float* __restrict__ stg,
                    int e0, int lm, int lane, int lhi, int llo)
{
    const int b   = nt >> 2;
    const int ntl = nt & 3;
    if (b == 0) {
        float* o = out + (size_t)e0 * OUT_W + 0 + 16 * ntl * 1;
        store_tile<1>(o, x2s_ + lm * X2D + 0, stg, acc, lane, lhi, llo);
    } else if (b == 1) {
        float* o = out + (size_t)e0 * OUT_W + 64 + 16 * ntl * 3;
        store_tile<3>(o, x2s_ + lm * X2D + 1, stg, acc, lane, lhi, llo);
    } else if (b == 2) {
        float* o = out + (size_t)e0 * OUT_W + 256 + 16 * ntl * 5;
        store_tile<5>(o, x2s_ + lm * X2D + 4, stg, acc, lane, lhi, llo);
    } else {
        float* o = out + (size_t)e0 * OUT_W + 576 + 16 * ntl * 7;
        store_tile<7>(o, x2s_ + lm * X2D + 9, stg, acc, lane, lhi, llo);
    }
}

template <bool PACKED>
__global__ __launch_bounds__(WAVES * 32)
void tp_fused_kernel(const float* __restrict__ x1,
                     const float* __restrict__ x2,
                     const float* __restrict__ W,
                     const f4* __restrict__ packB,
                     float* __restrict__ out, int E)
{
    __shared__ float x1s[EPB * PITCH];   // 33792 B; reused as store staging later
    __shared__ float x2s[EPB * X2D];     //  4096 B (pre-scaled by CG norm)

    const int tid  = threadIdx.x;
    const int eblk = blockIdx.x * EPB;

    const float r3 = 0.57735026918962576f;
    const float r5 = 0.44721359549995794f;
    const float r7 = 0.37796447300922722f;
    const float nfs[X2D] = {1.f, r3, r3, r3, r5, r5, r5, r5, r5,
                            r7, r7, r7, r7, r7, r7, r7};

    // ---- cooperative stage of x1 / x2 tiles (coalesced float4) ----
    {
        const int nvec = EPB * (U_DIM / 4);          // 2048 float4
        for (int v = tid; v < nvec; v += blockDim.x) {
            const int row = v >> 5;
            const int c4  = v & 31;
            f4 val = {0.f, 0.f, 0.f, 0.f};
            const int e = eblk + row;
            if (e < E) val = *(const f4*)(x1 + (size_t)e * U_DIM + c4 * 4);
            *(f4*)(&x1s[row * PITCH + c4 * 4]) = val;
        }
        const int nvec2 = EPB * (X2D / 4);           // 256 float4
        for (int v = tid; v < nvec2; v += blockDim.x) {
            const int row = v >> 2;
            const int c4  = v & 3;
            f4 val = {0.f, 0.f, 0.f, 0.f};
            const int e = eblk + row;
            if (e < E) val = *(const f4*)(x2 + (size_t)e * X2D + c4 * 4);
            val.x *= nfs[c4 * 4 + 0];
            val.y *= nfs[c4 * 4 + 1];
            val.z *= nfs[c4 * 4 + 2];
            val.w *= nfs[c4 * 4 + 3];
            *(f4*)(&x2s[row * X2D + c4 * 4]) = val;
        }
    }
    __syncthreads();

    const int wave = tid >> 5;
    const int lane = tid & 31;
    const int e0   = eblk + wave * EPW;              // wave-uniform
    const bool active = (e0 < E);

    const int lm  = wave * EPW;
    const int lhi = lane >> 4;
    const int llo = lane & 15;
    const int kk  = lhi * 2;

    // ---- hoist all A fragments into registers (reused by all 16 N-tiles) ----
    v2f af[KSTEPS];
    if (active) {
        #pragma unroll
        for (int ks = 0; ks < KSTEPS; ++ks) {
            const float* ap = &x1s[(lm + llo) * PITCH + ks * 4 + kk];
            af[ks].x = ap[0];
            af[ks].y = ap[1];
        }
    }
    __syncthreads();                 // x1s now dead -> becomes store staging
    if (!active) return;             // whole wave exits together

    float* stg = x1s + wave * STG_FLOATS;            // 7424 B per wave, 16B aligned

    // ---- B prefetch pipeline (PACKED path): 8-slot rotating buffer ----
    const f4* Bp = packB + lane;
    f4 buf[PF];
    if (PACKED) {
        #pragma unroll
        for (int i = 0; i < PF; ++i) buf[i] = Bp[(size_t)i * 32];
    }

    #pragma unroll 1
    for (int nt = 0; nt < 15; ++nt) {
        v8f acc = {};
        if (PACKED) gemm_tile<true>(acc, af, buf, Bp, nt);
        else        gemm_tile_direct(acc, af, W, nt, llo, kk);
        store_dispatch(nt, acc, out, x2s, stg, e0, lm, lane, lhi, llo);
    }
    {   // peeled last N-tile: no prefetch -> no clamp, no OOB
        v8f acc = {};
        if (PACKED) gemm_tile<false>(acc, af, buf, Bp, 15);
        else        gemm_tile_direct(acc, af, W, 15, llo, kk);
        store_dispatch(15, acc, out, x2s, stg, e0, lm, lane, lhi, llo);
    }
}

extern "C" void kernel_launch(void* const* d_in, const int* in_sizes, int n_in,
                              void* d_out, int out_size, void* d_ws, size_t ws_size,
                              hipStream_t stream) {
    const float* x1 = (const float*)d_in[0];   // [E,128]
    const float* x2 = (const float*)d_in[1];   // [E,16]
    const float* W  = (const float*)d_in[2];   // [4,128,64]
    float* out = (float*)d_out;                // [E,1024]

    const int E      = in_sizes[0] / U_DIM;
    const int tiles  = (E + EPW - 1) / EPW;
    const int blocks = (tiles + WAVES - 1) / WAVES;

    if (ws_size >= PACK_BYTES) {
        w_pack_kernel<<<(16 * KS2 * 32 + 255) / 256, 256, 0, stream>>>(W, (float*)d_ws);
        tp_fused_kernel<true><<<blocks, WAVES * 32, 0, stream>>>(
            x1, x2, W, (const f4*)d_ws, out, E);
    } else {
        tp_fused_kernel<false><<<blocks, WAVES * 32, 0, stream>>>(
            x1, x2, W, nullptr, out, E);
    }
}